// SMPL_18829136626391
// MI455X (gfx1250) — compile-verified
//
#include <hip/hip_runtime.h>
#include <hip/hip_bf16.h>

#define B_SZ 512
#define NJ   24
#define V_N  6890
#define V3   (V_N * 3)
#define NB_  10
#define NPF  207

typedef __attribute__((ext_vector_type(2))) float v2f;
typedef __attribute__((ext_vector_type(8))) float v8f;

__device__ __constant__ int c_par[NJ] = {
    -1, 0, 0, 0, 1, 2, 3, 4, 5, 6, 7, 8, 9, 9, 9, 12, 13, 14, 16, 17, 18, 19, 20, 21};

__device__ __forceinline__ v8f wmma4(v2f a, v2f b, v8f c) {
  // D = A(16x4,f32) * B(4x16,f32) + C(16x16,f32)
  return __builtin_amdgcn_wmma_f32_16x16x4_f32(false, a, false, b, (short)0, c,
                                               false, false);
}

// ---------------------------------------------------------------------------
// Kernel A: Rodrigues per (batch, joint). Writes Rs (B x 24 x 9) and
// pose_feature (B x 207) = (Rs[:,1:] - I) flattened.
// ---------------------------------------------------------------------------
__global__ __launch_bounds__(256) void k_rodrigues(
    const float* __restrict__ theta, float* __restrict__ Rs,
    float* __restrict__ pf) {
  int idx = blockIdx.x * blockDim.x + threadIdx.x;
  if (idx >= B_SZ * NJ) return;
  int b = idx / NJ, j = idx % NJ;
  float t0 = theta[b * NJ * 3 + j * 3 + 0];
  float t1 = theta[b * NJ * 3 + j * 3 + 1];
  float t2 = theta[b * NJ * 3 + j * 3 + 2];
  const float eps = 1e-8f;
  float a0 = t0 + eps, a1 = t1 + eps, a2 = t2 + eps;
  float angle = sqrtf(a0 * a0 + a1 * a1 + a2 * a2);
  float inv = 1.0f / angle;
  float nx = t0 * inv, ny = t1 * inv, nz = t2 * inv;
  float h = 0.5f * angle;
  float sh = sinf(h), ch = cosf(h);
  float qw = ch, qx = nx * sh, qy = ny * sh, qz = nz * sh;
  float qn = rsqrtf(qw * qw + qx * qx + qy * qy + qz * qz);
  qw *= qn; qx *= qn; qy *= qn; qz *= qn;
  float w2 = qw * qw, x2 = qx * qx, y2 = qy * qy, z2 = qz * qz;
  float wx = qw * qx, wy = qw * qy, wz = qw * qz;
  float xy = qx * qy, xz = qx * qz, yz = qy * qz;
  float R[9];
  R[0] = w2 + x2 - y2 - z2; R[1] = 2.f * (xy - wz);     R[2] = 2.f * (wy + xz);
  R[3] = 2.f * (wz + xy);   R[4] = w2 - x2 + y2 - z2;   R[5] = 2.f * (yz - wx);
  R[6] = 2.f * (xz - wy);   R[7] = 2.f * (wx + yz);     R[8] = w2 - x2 - y2 + z2;
  float* rb = Rs + (size_t)b * (NJ * 9) + j * 9;
#pragma unroll
  for (int e = 0; e < 9; e++) rb[e] = R[e];
  if (j >= 1) {
    float* pb = pf + (size_t)b * NPF + (j - 1) * 9;
#pragma unroll
    for (int e = 0; e < 9; e++)
      pb[e] = R[e] - ((e == 0 || e == 4 || e == 8) ? 1.f : 0.f);
  }
}

// ---------------------------------------------------------------------------
// Kernel B: Jdirs[k,j,c] = sum_v shapedirs[k, v*3+c] * J_reg[v,j]
//           J0[j,c]      = sum_v v_template[v,c]     * J_reg[v,j]
// One block per (j,c), 256-thread reduction over V.
// ---------------------------------------------------------------------------
__global__ __launch_bounds__(256) void k_jdirs(
    const float* __restrict__ vtpl, const float* __restrict__ sdirs,
    const float* __restrict__ jreg, float* __restrict__ Jdirs,
    float* __restrict__ J0) {
  const int j = blockIdx.x / 3, c = blockIdx.x % 3;
  const int tid = threadIdx.x;
  float acc[NB_ + 1];
#pragma unroll
  for (int k = 0; k <= NB_; k++) acc[k] = 0.f;
  for (int v = tid; v < V_N; v += 256) {
    float r = jreg[(size_t)v * NJ + j];
    acc[NB_] += vtpl[v * 3 + c] * r;
#pragma unroll
    for (int k = 0; k < NB_; k++)
      acc[k] += sdirs[(size_t)k * V3 + v * 3 + c] * r;
  }
  __shared__ float red[256];
  for (int k = 0; k <= NB_; k++) {
    red[tid] = acc[k];
    __syncthreads();
    for (int s = 128; s > 0; s >>= 1) {
      if (tid < s) red[tid] += red[tid + s];
      __syncthreads();
    }
    if (tid == 0) {
      if (k < NB_) Jdirs[k * (NJ * 3) + j * 3 + c] = red[0];
      else         J0[j * 3 + c] = red[0];
    }
    __syncthreads();
  }
}

// ---------------------------------------------------------------------------
// Kernel C: per-batch joints + kinematic chain -> A (B x 24 x 12, rows of 3x4)
// One wave per batch; chain walked serially by lane 0 in LDS.
// ---------------------------------------------------------------------------
__global__ __launch_bounds__(32) void k_rigid(
    const float* __restrict__ beta, const float* __restrict__ Rs,
    const float* __restrict__ Jdirs, const float* __restrict__ J0,
    float* __restrict__ Amat) {
  const int b = blockIdx.x;
  const int j = threadIdx.x;
  __shared__ float sJ[NJ][3];
  __shared__ float sM[NJ][12];
  if (j < NJ) {
    float bb[NB_];
#pragma unroll
    for (int k = 0; k < NB_; k++) bb[k] = beta[b * NB_ + k];
#pragma unroll
    for (int c = 0; c < 3; c++) {
      float acc = J0[j * 3 + c];
#pragma unroll
      for (int k = 0; k < NB_; k++) acc += bb[k] * Jdirs[k * (NJ * 3) + j * 3 + c];
      sJ[j][c] = acc;
    }
  }
  __syncthreads();
  if (j < NJ) {
    float R[9];
#pragma unroll
    for (int e = 0; e < 9; e++) R[e] = Rs[(size_t)b * (NJ * 9) + j * 9 + e];
    float t0, t1, t2;
    if (j == 0) {
      // root_rot = R @ diag(1,-1,-1): negate columns 1 and 2
      R[1] = -R[1]; R[2] = -R[2]; R[4] = -R[4];
      R[5] = -R[5]; R[7] = -R[7]; R[8] = -R[8];
      t0 = sJ[0][0]; t1 = sJ[0][1]; t2 = sJ[0][2];
    } else {
      int p = c_par[j];
      t0 = sJ[j][0] - sJ[p][0];
      t1 = sJ[j][1] - sJ[p][1];
      t2 = sJ[j][2] - sJ[p][2];
    }
    sM[j][0] = R[0]; sM[j][1] = R[1]; sM[j][2]  = R[2]; sM[j][3]  = t0;
    sM[j][4] = R[3]; sM[j][5] = R[4]; sM[j][6]  = R[5]; sM[j][7]  = t1;
    sM[j][8] = R[6]; sM[j][9] = R[7]; sM[j][10] = R[8]; sM[j][11] = t2;
  }
  __syncthreads();
  if (j == 0) {
    for (int i = 1; i < NJ; i++) {
      const int p = c_par[i];
      float G[12], L[12], N_[12];
      for (int e = 0; e < 12; e++) { G[e] = sM[p][e]; L[e] = sM[i][e]; }
#pragma unroll
      for (int r = 0; r < 3; r++) {
#pragma unroll
        for (int c = 0; c < 3; c++)
          N_[r * 4 + c] = G[r * 4 + 0] * L[0 * 4 + c] +
                          G[r * 4 + 1] * L[1 * 4 + c] +
                          G[r * 4 + 2] * L[2 * 4 + c];
        N_[r * 4 + 3] = G[r * 4 + 0] * L[3] + G[r * 4 + 1] * L[7] +
                        G[r * 4 + 2] * L[11] + G[r * 4 + 3];
      }
      for (int e = 0; e < 12; e++) sM[i][e] = N_[e];
    }
  }
  __syncthreads();
  if (j < NJ) {
    float G[12];
    for (int e = 0; e < 12; e++) G[e] = sM[j][e];
    float jx = sJ[j][0], jy = sJ[j][1], jz = sJ[j][2];
    float* o = Amat + (size_t)b * (NJ * 12) + j * 12;
#pragma unroll
    for (int r = 0; r < 3; r++) {
      float ib = G[r * 4 + 0] * jx + G[r * 4 + 1] * jy + G[r * 4 + 2] * jz;
      o[r * 4 + 0] = G[r * 4 + 0];
      o[r * 4 + 1] = G[r * 4 + 1];
      o[r * 4 + 2] = G[r * 4 + 2];
      o[r * 4 + 3] = G[r * 4 + 3] - ib;
    }
  }
}

// ---------------------------------------------------------------------------
// Kernel D (WMMA): fused shape/pose GEMM + LBS.
// One wave per (16-batch x 16-vertex) tile. Three f32 16x16 accumulators
// (x,y,z). K = 12 (beta, padded) + 208 (pose_feature, padded) via
// V_WMMA_F32_16X16X4_F32. Then blend against LDS-staged A matrices.
// ---------------------------------------------------------------------------
__global__ __launch_bounds__(32) void k_lbs(
    const float* __restrict__ beta, const float* __restrict__ pf,
    const float* __restrict__ vtpl, const float* __restrict__ sdirs,
    const float* __restrict__ pdirs, const float* __restrict__ Amat,
    const float* __restrict__ wts, float* __restrict__ verts) {
  __shared__ float sA[16 * NJ * 12];  // 18 KB: A for the 16 batches of tile
  const int lane = threadIdx.x;
  const int n = lane & 15;       // tile column (vertex) / A-operand row
  const int half = lane >> 4;    // K-half select
  const int kh = half * 2;
  const int vt = blockIdx.x * 16;
  const int bt = blockIdx.y * 16;
  const int vg = vt + n;
  const bool vok = vg < V_N;
  const int vc = vok ? vg : (V_N - 1);  // clamp for loads
  const int col = vc * 3;
  const int ab = bt + n;  // batch row feeding A operand

  // Cooperative stage of 16 contiguous A matrices (16*288 floats).
  const float* Asrc = Amat + (size_t)bt * (NJ * 12);
  for (int i = lane; i < 16 * NJ * 12; i += 32) sA[i] = Asrc[i];

  v8f ax = {}, ay = {}, az = {};

  // ---- shape GEMM: K = 10, padded to 12 (branchless; a=0 kills padding) ----
#pragma unroll
  for (int k0 = 0; k0 < 12; k0 += 4) {
    const int ka = k0 + kh;
    const int kc0 = ka < NB_ - 1 ? ka : NB_ - 1;
    const int kc1 = ka + 1 < NB_ ? ka + 1 : NB_ - 1;
    v2f a, bx, by, bz;
    float va0 = beta[ab * NB_ + kc0];
    float va1 = beta[ab * NB_ + kc1];
    a.x = (ka < NB_) ? va0 : 0.f;
    a.y = (ka + 1 < NB_) ? va1 : 0.f;
    const float* p0 = sdirs + (size_t)kc0 * V3 + col;
    const float* p1 = sdirs + (size_t)kc1 * V3 + col;
    bx.x = p0[0]; by.x = p0[1]; bz.x = p0[2];
    bx.y = p1[0]; by.y = p1[1]; bz.y = p1[2];
    ax = wmma4(a, bx, ax);
    ay = wmma4(a, by, ay);
    az = wmma4(a, bz, az);
  }

  // ---- pose GEMM: K = 207, padded to 208 ----
  for (int k0 = 0; k0 < 208; k0 += 4) {
    const int ka = k0 + kh;
    const int kc0 = ka < NPF - 1 ? ka : NPF - 1;
    const int kc1 = ka + 1 < NPF ? ka + 1 : NPF - 1;
    v2f a, bx, by, bz;
    float va0 = pf[(size_t)ab * NPF + kc0];
    float va1 = pf[(size_t)ab * NPF + kc1];
    a.x = (ka < NPF) ? va0 : 0.f;
    a.y = (ka + 1 < NPF) ? va1 : 0.f;
    const float* p0 = pdirs + (size_t)kc0 * V3 + col;
    const float* p1 = pdirs + (size_t)kc1 * V3 + col;
    bx.x = p0[0]; by.x = p0[1]; bz.x = p0[2];
    bx.y = p1[0]; by.y = p1[1]; bz.y = p1[2];
    ax = wmma4(a, bx, ax);
    ay = wmma4(a, by, ay);
    az = wmma4(a, bz, az);
  }

  // v_template offset + per-vertex skinning weights
  const float t0 = vtpl[col], t1 = vtpl[col + 1], t2 = vtpl[col + 2];
  float wj[NJ];
#pragma unroll
  for (int j = 0; j < NJ; j++) wj[j] = wts[(size_t)vc * NJ + j];

  __syncthreads();  // sA visible

  // LBS blend: lane holds vertex 'vg', 8 batch rows (C/D layout: VGPR r ->
  // row r (lanes 0-15) or r+8 (lanes 16-31)).
#pragma unroll
  for (int r = 0; r < 8; r++) {
    const int m = r + half * 8;
    const float x = ax[r] + t0, y = ay[r] + t1, z = az[r] + t2;
    float o0 = 0.f, o1 = 0.f, o2 = 0.f;
    const float* Ab = &sA[m * NJ * 12];
    for (int j = 0; j < NJ; j++) {
      const float* Aj = Ab + j * 12;
      const float ww = wj[j];
      o0 += ww * (Aj[0] * x + Aj[1] * y + Aj[2]  * z + Aj[3]);
      o1 += ww * (Aj[4] * x + Aj[5] * y + Aj[6]  * z + Aj[7]);
      o2 += ww * (Aj[8] * x + Aj[9] * y + Aj[10] * z + Aj[11]);
    }
    if (vok) {
      size_t o = ((size_t)(bt + m) * V_N + vg) * 3;
      verts[o] = o0; verts[o + 1] = o1; verts[o + 2] = o2;
    }
  }
}

// ---------------------------------------------------------------------------
// Kernel E: joints[b,j,:] = sum_v verts[b,v,:] * joint_regressor[v,j]
// ---------------------------------------------------------------------------
__global__ __launch_bounds__(256) void k_joints(
    const float* __restrict__ verts, const float* __restrict__ jr,
    float* __restrict__ joints) {
  const int b = blockIdx.x, j = blockIdx.y;
  const int tid = threadIdx.x;
  float a0 = 0.f, a1 = 0.f, a2 = 0.f;
  for (int v = tid; v < V_N; v += 256) {
    float wv = jr[(size_t)v * 19 + j];
    const float* p = verts + ((size_t)b * V_N + v) * 3;
    a0 += p[0] * wv; a1 += p[1] * wv; a2 += p[2] * wv;
  }
  __shared__ float red[3][256];
  red[0][tid] = a0; red[1][tid] = a1; red[2][tid] = a2;
  __syncthreads();
  for (int s = 128; s > 0; s >>= 1) {
    if (tid < s) {
      red[0][tid] += red[0][tid + s];
      red[1][tid] += red[1][tid + s];
      red[2][tid] += red[2][tid + s];
    }
    __syncthreads();
  }
  if (tid == 0) {
    float* o = joints + (size_t)b * 57 + j * 3;
    o[0] = red[0][0]; o[1] = red[1][0]; o[2] = red[2][0];
  }
}

// ---------------------------------------------------------------------------
extern "C" void kernel_launch(void* const* d_in, const int* in_sizes, int n_in,
                              void* d_out, int out_size, void* d_ws,
                              size_t ws_size, hipStream_t stream) {
  (void)in_sizes; (void)n_in; (void)out_size; (void)ws_size;
  const float* beta   = (const float*)d_in[0];
  const float* theta  = (const float*)d_in[1];
  const float* vtpl   = (const float*)d_in[2];
  const float* sdirs  = (const float*)d_in[3];
  const float* pdirs  = (const float*)d_in[4];
  const float* jreg   = (const float*)d_in[5];
  const float* jntreg = (const float*)d_in[6];
  const float* wts    = (const float*)d_in[7];
  float* out = (float*)d_out;
  float* ws  = (float*)d_ws;

  // ws layout (floats): pf | Rs | A | Jdirs | J0  (~1.46 MB total)
  float* pf    = ws;
  float* Rs    = pf + (size_t)B_SZ * NPF;
  float* Amat  = Rs + (size_t)B_SZ * NJ * 9;
  float* Jdirs = Amat + (size_t)B_SZ * NJ * 12;
  float* J0    = Jdirs + NB_ * NJ * 3;

  k_rodrigues<<<(B_SZ * NJ + 255) / 256, 256, 0, stream>>>(theta, Rs, pf);
  k_jdirs<<<NJ * 3, 256, 0, stream>>>(vtpl, sdirs, jreg, Jdirs, J0);
  k_rigid<<<B_SZ, 32, 0, stream>>>(beta, Rs, Jdirs, J0, Amat);
  dim3 g((V_N + 15) / 16, B_SZ / 16);
  k_lbs<<<g, 32, 0, stream>>>(beta, pf, vtpl, sdirs, pdirs, Amat, wts, out);
  dim3 gj(B_SZ, 19);
  k_joints<<<gj, 256, 0, stream>>>(out, jntreg, out + (size_t)B_SZ * V_N * 3);
}